// DCNv2Pooling_43550968381669
// MI455X (gfx1250) — compile-verified
//
#include <hip/hip_runtime.h>

// DCNv2 deformable PS-ROI pooling for MI455X (gfx1250).
// Pass 1: NCHW -> NHWC transpose into workspace, staged through LDS with CDNA5
//         async global->LDS copies (ASYNCcnt) -- 268 MB of streaming traffic.
// Pass 2: one block per ROI; 4 bins x 64 channel-quads in flight; float4 (B128)
//         coalesced gathers from NHWC rows; results staged in LDS and written
//         out as one linear 50 KB slab per ROI (pure streaming stores).
// Memory-bound op (~1.6 GFLOP vs ~320 MB HBM traffic); the 134 MB transposed
// feature map fits in the 192 MB L2, so pass-2 gathers are L2-resident.

#define SCALE     0.0625f
#define POOLED    7
#define TRANS_STD 0.1f

static constexpr int B_ = 8, C_ = 256, H_ = 128, W_ = 128;
static constexpr int HW_ = H_ * W_;
static constexpr int BINS_ = POOLED * POOLED;     // 49
static constexpr int OUT_PER_K_ = BINS_ * C_;     // 12544 floats per ROI

#if defined(__HIP_DEVICE_COMPILE__) && __has_builtin(__builtin_amdgcn_global_load_async_to_lds_b32)
#define USE_ASYNC_LDS 1
#endif

#define AS_GLOBAL __attribute__((address_space(1)))
#define AS_LDS    __attribute__((address_space(3)))

// ---------------------------------------------------------------------------
// Kernel 1: NCHW -> NHWC transpose via 32x32 LDS tiles.
// Reads coalesced along W, writes coalesced along C. Tile fill uses CDNA5
// async global->LDS copies (per-lane LDS scatter addresses, ASYNCcnt).
// ---------------------------------------------------------------------------
__global__ __launch_bounds__(256) void nchw_to_nhwc(const float* __restrict__ in,
                                                    float* __restrict__ out) {
    __shared__ float tile[32][33];  // +1 pad: conflict-free transposed reads

    const int x0 = blockIdx.x * 32;       // W tile origin
    const int c0 = blockIdx.y * 32;       // C tile origin
    const int by = blockIdx.z;            // flattened b*H + y
    const int tx = threadIdx.x;           // 0..31
    const int ty = threadIdx.y;           // 0..7

    const float* src = in + (size_t)(by / H_) * C_ * HW_ + (size_t)(by % H_) * W_;

#pragma unroll
    for (int i = 0; i < 4; ++i) {
        const int cl = ty + 8 * i;
        const float* g = src + (size_t)(c0 + cl) * HW_ + (x0 + tx);
#ifdef USE_ASYNC_LDS
        __builtin_amdgcn_global_load_async_to_lds_b32(
            (AS_GLOBAL int*)g,
            (AS_LDS int*)&tile[cl][tx],
            0, 0);
#else
        tile[cl][tx] = *g;
#endif
    }
#ifdef USE_ASYNC_LDS
#if __has_builtin(__builtin_amdgcn_s_wait_asynccnt)
    __builtin_amdgcn_s_wait_asynccnt(0);
#else
    asm volatile("s_wait_asynccnt 0" ::: "memory");
#endif
#endif
    __syncthreads();

    float* dst = out + (size_t)by * W_ * C_;
#pragma unroll
    for (int i = 0; i < 4; ++i) {
        const int xl = ty + 8 * i;
        dst[(size_t)(x0 + xl) * C_ + (c0 + tx)] = tile[tx][xl];
    }
}

// ---------------------------------------------------------------------------
// Kernel 2 (NHWC fast path): one 256-thread block (8 wave32) per ROI.
// tid = {bin subgroup (4) x channel quad (64)}. Gathers are float4/B128,
// 512 B per wave-load. Results staged in LDS, final store is one linear
// 50176 B slab per ROI.
// ---------------------------------------------------------------------------
__global__ __launch_bounds__(256) void dpool_nhwc(const float* __restrict__ feat,
                                                  const float* __restrict__ rois,
                                                  const float* __restrict__ offset,
                                                  float* __restrict__ out) {
    __shared__ __align__(16) float sout[OUT_PER_K_];   // 50176 B

    const int k    = blockIdx.x;
    const int tid  = threadIdx.x;
    const int grp  = tid >> 6;        // 0..3 : bin subgroup (2 waves each)
    const int lane = tid & 63;        // channel quad id
    const int c4   = lane * 4;

    // --- per-ROI geometry (uniform per block); rintf == jnp.round (RNE) ---
    const float* r = rois + (size_t)k * 5;
    const int   b   = (int)r[0];
    const float rsw = rintf(r[1]) * SCALE - 0.5f;
    const float rsh = rintf(r[2]) * SCALE - 0.5f;
    const float rew = (rintf(r[3]) + 1.0f) * SCALE - 0.5f;
    const float reh = (rintf(r[4]) + 1.0f) * SCALE - 0.5f;
    const float rw  = fmaxf(rew - rsw, 0.1f);
    const float rh  = fmaxf(reh - rsh, 0.1f);
    const float bw  = rw / (float)POOLED;
    const float bh  = rh / (float)POOLED;
    const float sw  = bw * 0.25f;     // SAMPLES = 4
    const float sh  = bh * 0.25f;

    const float* fb = feat + (size_t)b * HW_ * C_;

    for (int it = 0; it < (BINS_ + 3) / 4; ++it) {
        const int pp = it * 4 + grp;
        if (pp < BINS_) {
            const int ph = pp / POOLED;
            const int pw = pp % POOLED;
            // PART == POOLED -> part index is just (ph, pw)
            const float tx = offset[(size_t)(k * 2 + 0) * BINS_ + pp] * TRANS_STD;
            const float ty = offset[(size_t)(k * 2 + 1) * BINS_ + pp] * TRANS_STD;
            const float wstart = (float)pw * bw + rsw + tx * rw;
            const float hstart = (float)ph * bh + rsh + ty * rh;

            float4 acc = make_float4(0.f, 0.f, 0.f, 0.f);
            float  cnt = 0.0f;

#pragma unroll
            for (int s = 0; s < 16; ++s) {
                const float w = wstart + (float)(s & 3) * sw;
                const float h = hstart + (float)(s >> 2) * sh;
                // uniform within the bin's 2 waves -> uniform branch
                if (!((w >= -0.5f) & (w <= (float)W_ - 0.5f) &
                      (h >= -0.5f) & (h <= (float)H_ - 0.5f)))
                    continue;
                const float wc = fminf(fmaxf(w, 0.0f), (float)W_ - 1.0f);
                const float hc = fminf(fmaxf(h, 0.0f), (float)H_ - 1.0f);
                const int x1 = (int)floorf(wc);
                const int y1 = (int)floorf(hc);
                const int x2 = (int)ceilf(wc);
                const int y2 = (int)ceilf(hc);
                const float dx = wc - (float)x1;
                const float dy = hc - (float)y1;

                const float4 v11 = *(const float4*)(fb + (size_t)(y1 * W_ + x1) * C_ + c4);
                const float4 v12 = *(const float4*)(fb + (size_t)(y1 * W_ + x2) * C_ + c4);
                const float4 v21 = *(const float4*)(fb + (size_t)(y2 * W_ + x1) * C_ + c4);
                const float4 v22 = *(const float4*)(fb + (size_t)(y2 * W_ + x2) * C_ + c4);

                const float w11 = (1.0f - dx) * (1.0f - dy);
                const float w12 = dx * (1.0f - dy);
                const float w21 = (1.0f - dx) * dy;
                const float w22 = dx * dy;
                acc.x += w11 * v11.x + w12 * v12.x + w21 * v21.x + w22 * v22.x;
                acc.y += w11 * v11.y + w12 * v12.y + w21 * v21.y + w22 * v22.y;
                acc.z += w11 * v11.z + w12 * v12.z + w21 * v21.z + w22 * v22.z;
                acc.w += w11 * v11.w + w12 * v12.w + w21 * v21.w + w22 * v22.w;
                cnt += 1.0f;
            }

            // out layout per ROI: [c][pp]; stage in LDS for a linear writeout
            const bool nz = cnt > 0.0f;
            sout[(size_t)(c4 + 0) * BINS_ + pp] = nz ? acc.x / cnt : 0.0f;
            sout[(size_t)(c4 + 1) * BINS_ + pp] = nz ? acc.y / cnt : 0.0f;
            sout[(size_t)(c4 + 2) * BINS_ + pp] = nz ? acc.z / cnt : 0.0f;
            sout[(size_t)(c4 + 3) * BINS_ + pp] = nz ? acc.w / cnt : 0.0f;
        }
    }
    __syncthreads();

    // linear coalesced writeout: 12544 floats = 3136 float4 per ROI
    const float4* s4 = (const float4*)sout;
    float4* o4 = (float4*)(out + (size_t)k * OUT_PER_K_);
    for (int i = tid; i < OUT_PER_K_ / 4; i += 256)
        o4[i] = s4[i];
}

// ---------------------------------------------------------------------------
// Fallback (NCHW direct gather) if workspace is too small: one block per bin,
// thread = channel. Correct but uncoalesced gathers.
// ---------------------------------------------------------------------------
__global__ __launch_bounds__(256) void dpool_nchw(const float* __restrict__ feat,
                                                  const float* __restrict__ rois,
                                                  const float* __restrict__ offset,
                                                  float* __restrict__ out) {
    const int bin = blockIdx.x;
    const int k  = bin / BINS_;
    const int pp = bin % BINS_;
    const int ph = pp / POOLED;
    const int pw = pp % POOLED;
    const int c  = threadIdx.x;

    const float* r = rois + (size_t)k * 5;
    const int   b   = (int)r[0];
    const float rsw = rintf(r[1]) * SCALE - 0.5f;
    const float rsh = rintf(r[2]) * SCALE - 0.5f;
    const float rew = (rintf(r[3]) + 1.0f) * SCALE - 0.5f;
    const float reh = (rintf(r[4]) + 1.0f) * SCALE - 0.5f;
    const float rw  = fmaxf(rew - rsw, 0.1f);
    const float rh  = fmaxf(reh - rsh, 0.1f);
    const float bw  = rw / (float)POOLED;
    const float bh  = rh / (float)POOLED;
    const float sw  = bw * 0.25f;
    const float sh  = bh * 0.25f;
    const float tx = offset[(size_t)(k * 2 + 0) * BINS_ + pp] * TRANS_STD;
    const float ty = offset[(size_t)(k * 2 + 1) * BINS_ + pp] * TRANS_STD;
    const float wstart = (float)pw * bw + rsw + tx * rw;
    const float hstart = (float)ph * bh + rsh + ty * rh;

    const float* fb = feat + ((size_t)b * C_ + c) * HW_;

    float acc = 0.0f, cnt = 0.0f;
#pragma unroll
    for (int s = 0; s < 16; ++s) {
        const float w = wstart + (float)(s & 3) * sw;
        const float h = hstart + (float)(s >> 2) * sh;
        if (!((w >= -0.5f) & (w <= (float)W_ - 0.5f) &
              (h >= -0.5f) & (h <= (float)H_ - 0.5f)))
            continue;
        const float wc = fminf(fmaxf(w, 0.0f), (float)W_ - 1.0f);
        const float hc = fminf(fmaxf(h, 0.0f), (float)H_ - 1.0f);
        const int x1 = (int)floorf(wc);
        const int y1 = (int)floorf(hc);
        const int x2 = (int)ceilf(wc);
        const int y2 = (int)ceilf(hc);
        const float dx = wc - (float)x1;
        const float dy = hc - (float)y1;
        const float v11 = fb[(size_t)y1 * W_ + x1];
        const float v12 = fb[(size_t)y1 * W_ + x2];
        const float v21 = fb[(size_t)y2 * W_ + x1];
        const float v22 = fb[(size_t)y2 * W_ + x2];
        acc += (1.0f - dx) * (1.0f - dy) * v11 + dx * (1.0f - dy) * v12 +
               (1.0f - dx) * dy * v21 + dx * dy * v22;
        cnt += 1.0f;
    }
    out[((size_t)k * C_ + c) * BINS_ + pp] = (cnt > 0.0f) ? acc / cnt : 0.0f;
}

extern "C" void kernel_launch(void* const* d_in, const int* in_sizes, int n_in,
                              void* d_out, int out_size, void* d_ws, size_t ws_size,
                              hipStream_t stream) {
    const float* x      = (const float*)d_in[0];  // (B,C,H,W) f32
    const float* rois   = (const float*)d_in[1];  // (K,5) f32
    const float* offset = (const float*)d_in[2];  // (K,2,7,7) f32
    float*       out    = (float*)d_out;          // (K,C,7,7) f32

    const int K = in_sizes[1] / 5;
    const size_t need = (size_t)B_ * C_ * H_ * W_ * sizeof(float);

    if (ws_size >= need) {
        float* nhwc = (float*)d_ws;
        dim3 tb(32, 8);
        dim3 tg(W_ / 32, C_ / 32, B_ * H_);
        nchw_to_nhwc<<<tg, tb, 0, stream>>>(x, nhwc);
        dpool_nhwc<<<K, 256, 0, stream>>>(nhwc, rois, offset, out);
    } else {
        dpool_nchw<<<K * BINS_, 256, 0, stream>>>(x, rois, offset, out);
    }
}